// LegoConv2d_42863773614191
// MI455X (gfx1250) — compile-verified
//
#include <hip/hip_runtime.h>
#include <hip/hip_bf16.h>

typedef __attribute__((ext_vector_type(2))) float v2f;
typedef __attribute__((ext_vector_type(8))) float v8f;

#define IN_C    256
#define OUT_C   256
#define N_SPLIT 4
#define N_LEGO  64
#define BASIC_C 64
#define HIN     56
#define WIN     56
#define HOUT    19
#define WOUT    58
#define KDIM    192      // BASIC_C * 3 (kh)
#define NCOL    64       // padded N (cols 0 and 57..63 are zero)
#define XSP_STR 160      // floats per k-pair row: 128 data + 32 pad
                         // (160 mod 64 == 32 -> half-waves hit disjoint bank halves)
#define FS_W    64       // feat LDS row

__global__ __launch_bounds__(256, 1)
void lego_conv_fused(const float* __restrict__ x,
                     const float* __restrict__ lego,
                     const float* __restrict__ coef,
                     const float* __restrict__ comb,
                     float* __restrict__ out) {
    // B panel, k-pair interleaved: element (k, c) at (k>>1)*XSP_STR + 2c + (k&1)
    __shared__ float XsP[(KDIM / 2) * XSP_STR];    // 96*160*4 = 61,440 B
    __shared__ float Fs[N_SPLIT * N_LEGO * FS_W];  // 65,536 B : feat, all 4 splits
    __shared__ float cW[N_SPLIT * OUT_C];          //  4,096 B : selected coefficient
    __shared__ int   iW[N_SPLIT * OUT_C];          //  4,096 B : argmax lego index

    const int tid = threadIdx.x;
    const int b   = blockIdx.y;   // batch 0..31
    const int h   = blockIdx.x;   // output row 0..18

    // ---- Stage 0: per-(split, out-channel) argmax + coefficient ----------
    for (int p = tid; p < N_SPLIT * OUT_C; p += 256) {
        const float* cb = comb + p * N_LEGO;
        int   best = 0;
        float bv   = cb[0];
        #pragma unroll
        for (int n = 1; n < N_LEGO; ++n) {
            float v = cb[n];
            if (v > bv) { bv = v; best = n; }   // strict > keeps first max (jnp.argmax)
        }
        cW[p] = coef[p * N_LEGO + best];
        iW[p] = best;
    }

    const int lane = tid & 31;
    const int wave = tid >> 5;
    const int half = lane >> 4;   // 0: lanes 0-15, 1: lanes 16-31
    const int l16  = lane & 15;
    const int mt   = wave >> 1;        // M-tile 0..3 (feat channel / 16)
    const int nt0  = (wave & 1) * 2;   // N-tiles {nt0, nt0+1}

    // A fragment source: A[m][k] = lego[m*192 + k]; 16x4 fp32 layout:
    //   lanes 0-15 hold K = kb+0 / kb+1, lanes 16-31 hold K = kb+2 / kb+3
    const float* Arow = lego + (mt * 16 + l16) * KDIM + half * 2;

    for (int s = 0; s < N_SPLIT; ++s) {
        __syncthreads();  // previous split's consumers of XsP/Fs are done

        // ---- Stage 1: stage B panel into LDS (pair-interleaved, zero-padded)
        for (int j = tid; j < KDIM * NCOL; j += 256) {
            int   k = j >> 6;          // 0..191
            int   w = j & 63;          // padded output col 0..63
            float val = 0.0f;
            if (w >= 1 && w <= WIN) {  // cols 0 and 57..63 stay zero
                int ci = k / 3;
                int kh = k - ci * 3;
                int r  = 3 * h + kh - 1;            // input row (stride 3, pad 1)
                if (r >= 0)                          // h==0,kh==0 -> padded row -1
                    val = x[(((size_t)b * IN_C + s * BASIC_C + ci) * HIN + r) * WIN + (w - 1)];
            }
            XsP[(k >> 1) * XSP_STR + (w << 1) + (k & 1)] = val;
        }
        __syncthreads();

        // ---- Stage 2: feat(64x64) = A(64x192) * B(192x64), fp32 WMMA -----
        v8f acc0 = {0.f, 0.f, 0.f, 0.f, 0.f, 0.f, 0.f, 0.f};
        v8f acc1 = {0.f, 0.f, 0.f, 0.f, 0.f, 0.f, 0.f, 0.f};
        #pragma unroll 4
        for (int kb = 0; kb < KDIM; kb += 4) {
            v2f a = *(const v2f*)(Arow + kb);   // 8B-aligned global load (L1-resident)
            // lanes 0-15 read k-pair kb/2, lanes 16-31 read pair kb/2+1:
            const float* brow = &XsP[((kb >> 1) + half) * XSP_STR + (l16 << 1)];
            v2f b0 = *(const v2f*)(brow + nt0 * 32);        // single ds b64 load
            v2f b1 = *(const v2f*)(brow + nt0 * 32 + 32);
            acc0 = __builtin_amdgcn_wmma_f32_16x16x4_f32(
                       false, a, false, b0, (short)0, acc0, false, false);
            acc1 = __builtin_amdgcn_wmma_f32_16x16x4_f32(
                       false, a, false, b1, (short)0, acc1, false, false);
        }

        // ---- Stage 3: spill C fragments to LDS feat buffer ---------------
        // C/D layout: VGPR v holds row (v + 8*half), col = l16 (+ 16*nt)
        float* fbase = Fs + (s * N_LEGO + mt * 16 + half * 8) * FS_W;
        #pragma unroll
        for (int v = 0; v < 8; ++v) {
            fbase[v * FS_W + nt0       * 16 + l16] = acc0[v];
            fbase[v * FS_W + (nt0 + 1) * 16 + l16] = acc1[v];
        }
    }
    __syncthreads();

    // ---- Stage 4: out[b,o,h,w] = sum_s cW[s,o] * Fs[s][iW[s,o]][w] -------
    const int wl = tid & 63;   // output col
    const int o0 = tid >> 6;   // 0..3
    if (wl < WOUT) {
        for (int o = o0; o < OUT_C; o += 4) {
            float acc = 0.f;
            #pragma unroll
            for (int s = 0; s < N_SPLIT; ++s) {
                int n = iW[s * OUT_C + o];
                acc += cW[s * OUT_C + o] * Fs[(s * N_LEGO + n) * FS_W + wl];
            }
            out[(((size_t)b * OUT_C + o) * HOUT + h) * WOUT + wl] = acc;
        }
    }
}

extern "C" void kernel_launch(void* const* d_in, const int* in_sizes, int n_in,
                              void* d_out, int out_size, void* d_ws, size_t ws_size,
                              hipStream_t stream) {
    (void)in_sizes; (void)n_in; (void)d_ws; (void)ws_size; (void)out_size;
    const float* x    = (const float*)d_in[0];   // (32,256,56,56)
    const float* lego = (const float*)d_in[1];   // (64,64,3,1)
    const float* coef = (const float*)d_in[2];   // (4,256,64,1,1)
    const float* comb = (const float*)d_in[3];   // (4,256,64,1,1)
    float* out = (float*)d_out;                  // (32,256,19,58)

    dim3 grid(HOUT, 32, 1);   // (h, b)
    dim3 block(256, 1, 1);    // 8 wave32
    lego_conv_fused<<<grid, block, 0, stream>>>(x, lego, coef, comb, out);
}